// myModel_68796786147565
// MI455X (gfx1250) — compile-verified
//
#include <hip/hip_runtime.h>
#include <math.h>

// ---------------------------------------------------------------------------
// Problem constants (match reference)
// ---------------------------------------------------------------------------
#define NN   10000
#define K1   10
#define K2   25
#define DIMC 64
#define OUTC 128
#define NGR  64

typedef __attribute__((ext_vector_type(2))) float v2f;
typedef __attribute__((ext_vector_type(8))) float v8f;

// ---------------------------------------------------------------------------
// mean_rows<D,K>: out[r, :] = mean_{j<K} in[r*K + j, :]
// One wave (32 lanes) per row, each lane owns D/32 contiguous floats.
// blockDim = 256 -> 8 rows per block. Fully coalesced 256B-per-wave streams.
// ---------------------------------------------------------------------------
template <int D, int KN>
__global__ void mean_rows(const float* __restrict__ in,
                          float* __restrict__ out, int rows) {
    constexpr int VW = D / 32;
    const int lane = threadIdx.x & 31;
    const int row  = blockIdx.x * (blockDim.x >> 5) + (threadIdx.x >> 5);
    if (row >= rows) return;

    float acc[VW];
#pragma unroll
    for (int v = 0; v < VW; ++v) acc[v] = 0.0f;

    const float* base = in + (size_t)row * KN * D + lane * VW;
#pragma unroll 5
    for (int j = 0; j < KN; ++j) {
#pragma unroll
        for (int v = 0; v < VW; ++v) acc[v] += base[(size_t)j * D + v];
    }
    const float invk = 1.0f / (float)KN;
    float* o = out + (size_t)row * D + lane * VW;
#pragma unroll
    for (int v = 0; v < VW; ++v) o[v] = acc[v] * invk;
}

// ---------------------------------------------------------------------------
// sage_gemm_relu<KHALF>:
//   out[r, :] = relu( concat(xL[r,:], xR[r,:]) @ W + bias ),   W: (2*KHALF, 128)
// Grid: rows/16 blocks of 256 threads (8 waves). Block stages the 16x(2*KHALF)
// concat A-tile in LDS; wave w computes the 16x16 output tile at columns
// [16w, 16w+16) with V_WMMA_F32_16X16X4_F32 over KW = 2*KHALF.
// rows must be a multiple of 16 (EXEC stays all-ones for WMMA).
// ---------------------------------------------------------------------------
template <int KHALF>
__global__ void sage_gemm_relu(const float* __restrict__ xL,
                               const float* __restrict__ xR,
                               const float* __restrict__ W,
                               const float* __restrict__ bias,
                               float* __restrict__ out, int rows) {
    constexpr int KW  = 2 * KHALF;     // total K
    constexpr int LDA = KW + 2;        // padded LDS stride (even, kills conflicts)
    __shared__ float As[16 * LDA];

    const int tid  = threadIdx.x;
    const int row0 = blockIdx.x * 16;

    // Cooperative load of concat(xL, xR) tile: 16 rows x KW cols.
    for (int i = tid; i < 16 * KW; i += 256) {
        const int r = i / KW;
        const int c = i - r * KW;
        const float v = (c < KHALF)
                            ? xL[(size_t)(row0 + r) * KHALF + c]
                            : xR[(size_t)(row0 + r) * KHALF + (c - KHALF)];
        As[r * LDA + c] = v;
    }
    __syncthreads();

    const int lane = tid & 31;
    const int wave = tid >> 5;          // 0..7 -> output column tile
    const int col0 = wave * 16;
    const int m    = lane & 15;         // A row / B,C,D column within tile
    const int khi  = (lane >> 4) * 2;   // lanes 16-31 hold K+2, K+3

    v8f acc = {};
#pragma unroll 8
    for (int k = 0; k < KW; k += 4) {
        v2f a, b;
        a.x = As[m * LDA + k + khi];
        a.y = As[m * LDA + k + khi + 1];
        b.x = W[(size_t)(k + khi) * OUTC + col0 + m];
        b.y = W[(size_t)(k + khi + 1) * OUTC + col0 + m];
        // v_wmma_f32_16x16x4_f32: (neg_a, A, neg_b, B, c_mod, C, reuse_a, reuse_b)
        acc = __builtin_amdgcn_wmma_f32_16x16x4_f32(
            false, a, false, b, (short)0, acc, false, false);
    }

    const float bv    = bias[col0 + m];
    const int   rbase = row0 + ((lane >> 4) * 8);  // C/D: VGPR g -> M = g (+8)
#pragma unroll
    for (int g = 0; g < 8; ++g) {
        float v = acc[g] + bv;
        out[(size_t)(rbase + g) * OUTC + col0 + m] = v > 0.0f ? v : 0.0f;
    }
}

// ---------------------------------------------------------------------------
// Segment sum over sorted graph ids. One block per graph: binary-search the
// contiguous row range, then coalesced accumulation (no atomics needed).
// ---------------------------------------------------------------------------
__device__ __forceinline__ int lbound(const int* __restrict__ a, int n, int v) {
    int lo = 0, hi = n;
    while (lo < hi) {
        int mid = (lo + hi) >> 1;
        if (a[mid] < v) lo = mid + 1; else hi = mid;
    }
    return lo;
}

__global__ void segsum_kernel(const float* __restrict__ x,
                              const int* __restrict__ gid,
                              float* __restrict__ seg, int n) {
    const int g = blockIdx.x;
    const int c = threadIdx.x;            // 128 threads
    const int lo = lbound(gid, n, g);
    const int hi = lbound(gid, n, g + 1);
    float s = 0.0f;
    for (int r = lo; r < hi; ++r) s += x[(size_t)r * OUTC + c];
    seg[(size_t)g * OUTC + c] = s;
}

// ---------------------------------------------------------------------------
// Readout MLP: selu(seg@wr1+br1) -> selu(@wr2+br2) -> @wr3+br3. One block/graph.
// ---------------------------------------------------------------------------
__device__ __forceinline__ float selu_f(float x) {
    const float alpha = 1.6732632423543772f;
    const float scale = 1.0507009873554805f;
    return scale * (x > 0.0f ? x : alpha * (__expf(x) - 1.0f));
}

__global__ void readout_kernel(const float* __restrict__ seg,
                               const float* __restrict__ wr1, const float* __restrict__ br1,
                               const float* __restrict__ wr2, const float* __restrict__ br2,
                               const float* __restrict__ wr3, const float* __restrict__ br3,
                               float* __restrict__ out) {
    __shared__ float r1[35];
    __shared__ float r2[35];
    const int g = blockIdx.x;
    const int t = threadIdx.x;            // 64 threads, 35 active
    if (t < 35) {
        float s = br1[t];
        for (int k = 0; k < OUTC; ++k) s += seg[(size_t)g * OUTC + k] * wr1[k * 35 + t];
        r1[t] = selu_f(s);
    }
    __syncthreads();
    if (t < 35) {
        float s = br2[t];
        for (int k = 0; k < 35; ++k) s += r1[k] * wr2[k * 35 + t];
        r2[t] = selu_f(s);
    }
    __syncthreads();
    if (t == 0) {
        float s = br3[0];
        for (int k = 0; k < 35; ++k) s += r2[k] * wr3[k];
        out[g] = s;
    }
}

// ---------------------------------------------------------------------------
// Pipeline
// ---------------------------------------------------------------------------
extern "C" void kernel_launch(void* const* d_in, const int* in_sizes, int n_in,
                              void* d_out, int out_size, void* d_ws, size_t ws_size,
                              hipStream_t stream) {
    const float* h0  = (const float*)d_in[0];
    const float* h1  = (const float*)d_in[1];
    const float* h2  = (const float*)d_in[2];
    const float* w0  = (const float*)d_in[3];
    const float* b0  = (const float*)d_in[4];
    const float* w1  = (const float*)d_in[5];
    const float* b1  = (const float*)d_in[6];
    const float* wr1 = (const float*)d_in[7];
    const float* br1 = (const float*)d_in[8];
    const float* wr2 = (const float*)d_in[9];
    const float* br2 = (const float*)d_in[10];
    const float* wr3 = (const float*)d_in[11];
    const float* br3 = (const float*)d_in[12];
    const int*   gid = (const int*)d_in[13];

    float* ws = (float*)d_ws;
    size_t off = 0;
    float* mean2 = ws + off; off += (size_t)NN * K1 * DIMC;  // (100000, 64)
    float* mean1 = ws + off; off += (size_t)NN * DIMC;       // (10000, 64)
    float* a12   = ws + off; off += (size_t)NN * K1 * OUTC;  // (100000, 128)
    float* a01   = ws + off; off += (size_t)NN * OUTC;       // (10000, 128)
    float* mA12  = ws + off; off += (size_t)NN * OUTC;       // (10000, 128)
    float* out2  = ws + off; off += (size_t)NN * OUTC;       // (10000, 128)
    float* seg   = ws + off; off += (size_t)NGR * OUTC;      // (64, 128)

    // Round 0 aggregations (memory-bound; 640 MB dominant sweep of h2)
    mean_rows<DIMC, K2><<<(NN * K1) / 8, 256, 0, stream>>>(h2, mean2, NN * K1);
    mean_rows<DIMC, K1><<<NN / 8, 256, 0, stream>>>(h1, mean1, NN);

    // Round 0 GEMMs (fp32 WMMA, fused concat+bias+relu)
    sage_gemm_relu<DIMC><<<(NN * K1) / 16, 256, 0, stream>>>(h1, mean2, w0, b0, a12, NN * K1);
    sage_gemm_relu<DIMC><<<NN / 16, 256, 0, stream>>>(h0, mean1, w0, b0, a01, NN);

    // Round 1
    mean_rows<OUTC, K1><<<NN / 8, 256, 0, stream>>>(a12, mA12, NN);
    sage_gemm_relu<OUTC><<<NN / 16, 256, 0, stream>>>(a01, mA12, w1, b1, out2, NN);

    // Segment sum + readout
    segsum_kernel<<<NGR, OUTC, 0, stream>>>(out2, gid, seg, NN);
    readout_kernel<<<NGR, 64, 0, stream>>>(seg, wr1, br1, wr2, br2, wr3, br3,
                                           (float*)d_out);
}